// Sp_GAT_69612829934216
// MI455X (gfx1250) — compile-verified
//
#include <hip/hip_runtime.h>
#include <hip/hip_bf16.h>

#define D 96
#define NEG_SLOPE 0.01f

typedef __attribute__((ext_vector_type(2))) float v2f;
typedef __attribute__((ext_vector_type(8))) float v8f;

// ---------------------------------------------------------------------------
// H[n x 96] = X[n x 96] @ W[96 x 96], full fp32 via V_WMMA_F32_16X16X4_F32.
// 4 waves per block; each wave owns a 16-row strip and 6 column tiles (6x16=96).
// A-frag layout (ISA 7.12.2, 32-bit A 16x4): lane%16 = M, half-wave selects
// K pair {0,1} vs {2,3} across the 2 A VGPRs. B mirrored with N = lane%16.
// C/D: vgpr v -> row M = v + 8*(lane>=16), col N = lane%16.
// ---------------------------------------------------------------------------
__global__ __launch_bounds__(128) void gat_gemm_wmma(const float* __restrict__ X,
                                                     const float* __restrict__ W,
                                                     float* __restrict__ H, int n) {
    __shared__ float sW[D * D];        // 36 KB
    __shared__ float sX[4][16 * 97];   // per-wave 16x96 tile, padded stride 97
    const int tid  = threadIdx.x;
    const int lane = tid & 31;
    const int wid  = tid >> 5;

    // Stage W (row-major, W[k][c] at k*96+c) cooperatively.
    for (int i = tid; i < D * D; i += 128) sW[i] = W[i];

    // Stage this wave's 16 contiguous X rows (zero-fill OOB rows: WMMA needs
    // full EXEC, so we compute garbage-free and guard only the stores).
    const int rowBase = (blockIdx.x * 4 + wid) * 16;
    for (int i = lane; i < 16 * D; i += 32) {
        int r = i / D, c = i - r * D;
        int gr = rowBase + r;
        sX[wid][r * 97 + c] = (gr < n) ? X[(size_t)gr * D + c] : 0.0f;
    }
    __syncthreads();

    v8f acc[6];
#pragma unroll
    for (int t = 0; t < 6; ++t) acc[t] = (v8f)(0.0f);

    const int m  = lane & 15;
    const int hi = lane >> 4;               // half-wave selector
#pragma unroll 4
    for (int kt = 0; kt < D / 4; ++kt) {
        const int k = kt * 4 + hi * 2;
        v2f a;
        a.x = sX[wid][m * 97 + k];
        a.y = sX[wid][m * 97 + k + 1];
#pragma unroll
        for (int t = 0; t < 6; ++t) {
            v2f b;
            b.x = sW[k * D + t * 16 + m];
            b.y = sW[(k + 1) * D + t * 16 + m];
            acc[t] = __builtin_amdgcn_wmma_f32_16x16x4_f32(
                /*neg_a=*/false, a, /*neg_b=*/false, b,
                /*c_mod=*/(short)0, acc[t], /*reuse_a=*/false, /*reuse_b=*/false);
        }
    }

#pragma unroll
    for (int t = 0; t < 6; ++t) {
#pragma unroll
        for (int v = 0; v < 8; ++v) {
            int row = rowBase + v + hi * 8;
            if (row < n) H[(size_t)row * D + t * 16 + m] = acc[t][v];
        }
    }
}

// ---------------------------------------------------------------------------
// Per-node attention scalars: asrc[v]=h[v]·a[0:96], adst[v]=h[v]·a[96:192].
// One wave32 per node, coalesced reads, shuffle reduction.
// ---------------------------------------------------------------------------
__global__ __launch_bounds__(256) void gat_alpha(const float* __restrict__ H,
                                                 const float* __restrict__ a,
                                                 float* __restrict__ asrc,
                                                 float* __restrict__ adst, int n) {
    int node = blockIdx.x * 8 + (threadIdx.x >> 5);
    int lane = threadIdx.x & 31;
    if (node >= n) return;
    float s1 = 0.0f, s2 = 0.0f;
    for (int c = lane; c < D; c += 32) {
        float hv = H[(size_t)node * D + c];
        s1 += hv * a[c];
        s2 += hv * a[D + c];
    }
    for (int o = 16; o > 0; o >>= 1) {
        s1 += __shfl_xor(s1, o, 32);
        s2 += __shfl_xor(s2, o, 32);
    }
    if (lane == 0) { asrc[node] = s1; adst[node] = s2; }
}

// ---------------------------------------------------------------------------
// Per-edge weight e = exp(-leaky_relu(asrc[s]+adst[d])), accumulate rowsum[s].
// ---------------------------------------------------------------------------
__global__ void gat_edge(const int* __restrict__ src, const int* __restrict__ dst,
                         const float* __restrict__ asrc, const float* __restrict__ adst,
                         float* __restrict__ ew, float* __restrict__ rowsum, int e) {
    int i = blockIdx.x * blockDim.x + threadIdx.x;
    if (i >= e) return;
    int s = src[i], d = dst[i];
    float v = asrc[s] + adst[d];
    float lr = v > 0.0f ? v : NEG_SLOPE * v;
    float w = expf(-lr);
    ew[i] = w;
    atomicAdd(&rowsum[s], w);
}

// ---------------------------------------------------------------------------
// Scatter: acc[src[e]] += ew[e] * H[dst[e]].  Wave per edge; lanes stride the
// 96 features -> 3 coalesced 128B loads + 3 global f32 atomics per lane.
// L2-resident on MI455X (38 MB feature buffer << 192 MB L2).
// ---------------------------------------------------------------------------
__global__ __launch_bounds__(256) void gat_scatter(const int* __restrict__ src,
                                                   const int* __restrict__ dst,
                                                   const float* __restrict__ ew,
                                                   const float* __restrict__ H,
                                                   float* __restrict__ acc, int e) {
    int edge = blockIdx.x * 8 + (threadIdx.x >> 5);
    int lane = threadIdx.x & 31;
    if (edge >= e) return;
    int s = src[edge], d = dst[edge];
    float w = ew[edge];
    for (int c = lane; c < D; c += 32)
        atomicAdd(&acc[(size_t)s * D + c], w * H[(size_t)d * D + c]);
}

__global__ void gat_norm(const float* __restrict__ acc, const float* __restrict__ rowsum,
                         float* __restrict__ out, int total, int relu) {
    int i = blockIdx.x * blockDim.x + threadIdx.x;
    if (i >= total) return;
    float v = acc[i] / rowsum[i / D];
    if (relu) v = fmaxf(v, 0.0f);
    out[i] = v;
}

__global__ void fill_zero(float* __restrict__ p, int n) {
    int i = blockIdx.x * blockDim.x + threadIdx.x;
    if (i < n) p[i] = 0.0f;
}

// ---------------------------------------------------------------------------
// inputs: [0] edge_index (2*E int32), [1] x (N*96 f32), [2] W1, [3] a1,
//         [4] W2, [5] a2.  out: N*96 f32.
// ---------------------------------------------------------------------------
extern "C" void kernel_launch(void* const* d_in, const int* in_sizes, int n_in,
                              void* d_out, int out_size, void* d_ws, size_t ws_size,
                              hipStream_t stream) {
    const int*   edge_index = (const int*)d_in[0];
    const float* x  = (const float*)d_in[1];
    const float* W1 = (const float*)d_in[2];
    const float* a1 = (const float*)d_in[3];
    const float* W2 = (const float*)d_in[4];
    const float* a2 = (const float*)d_in[5];
    const int e = in_sizes[0] / 2;
    const int n = in_sizes[1] / D;
    const int* src = edge_index;
    const int* dst = edge_index + e;
    const int nd = n * D;

    float* ws      = (float*)d_ws;
    float* bufH    = ws;                    // n*D : W-transformed features
    float* bufAcc  = bufH   + (size_t)nd;   // n*D : scatter accumulator / layer-1 out
    float* asrc    = bufAcc + (size_t)nd;   // n
    float* adst    = asrc   + n;            // n
    float* rowsum  = adst   + n;            // n
    float* ew      = rowsum + n;            // e

    auto layer = [&](const float* hin, const float* W, const float* a,
                     float* hout, int relu) {
        gat_gemm_wmma<<<(n + 63) / 64, 128, 0, stream>>>(hin, W, bufH, n);
        gat_alpha   <<<(n + 7) / 8, 256, 0, stream>>>(bufH, a, asrc, adst, n);
        fill_zero   <<<(nd + 255) / 256, 256, 0, stream>>>(bufAcc, nd);
        fill_zero   <<<(n + 255) / 256, 256, 0, stream>>>(rowsum, n);
        gat_edge    <<<(e + 255) / 256, 256, 0, stream>>>(src, dst, asrc, adst, ew, rowsum, e);
        gat_scatter <<<(e + 7) / 8, 256, 0, stream>>>(src, dst, ew, bufH, bufAcc, e);
        gat_norm    <<<(nd + 255) / 256, 256, 0, stream>>>(bufAcc, rowsum, hout, nd, relu);
    };

    // Layer 1: x -> bufAcc (normalized in place; read before re-zero in layer 2)
    layer(x, W1, a1, bufAcc, /*relu=*/0);
    // Layer 2: bufAcc -> d_out with final ReLU
    layer(bufAcc, W2, a2, (float*)d_out, /*relu=*/1);
}